// SinkhornLayer_28999619183029
// MI455X (gfx1250) — compile-verified
//
#include <hip/hip_runtime.h>

// ---------------------------------------------------------------------------
// Sinkhorn (log domain), 8192x8192, eps=0.05, 50 iters, fp32 exact-range math.
//
// Row LSE pass: CDNA5 async copy (global_load_async_to_lds_b128, ASYNCcnt)
// double-buffers 16x32 cost tiles into LDS; A-fragments are read back with
// bank-conflict-free ds_load_b64 (pitch = 36 floats = 4*odd -> 64-bank perm),
// exp2 applied in VALU, and V_WMMA_F32_16X16X4_F32 against an all-ones B
// accumulates 16 row sums in the f32 accumulator (D column-replicated).
//
// Everything is sized for HBM bandwidth (23.3 TB/s): one cost-matrix read per
// reduction pass, coalesced 512B wave transfers, ~4 waves/SIMD with 1-chunk
// async prefetch depth.
// ---------------------------------------------------------------------------

typedef float v2f __attribute__((ext_vector_type(2)));
typedef float v8f __attribute__((ext_vector_type(8)));

constexpr int   NDIM    = 8192;
constexpr float EPS     = 0.05f;
constexpr float INV_EPS = 20.0f;
constexpr float LOG2E   = 1.4426950408889634f;
constexpr float K2      = INV_EPS * LOG2E;   // exp((g-c)/eps) == exp2(g*K2 - c*K2)

// row pass tiling
constexpr int TK        = 32;                // k-columns per staged tile
constexpr int PITCH     = 36;                // floats; 4*odd -> conflict-free A frags
constexpr int NCHUNK    = NDIM / TK;         // 256
constexpr int RPB       = 16;                // rows per block (one WMMA M-tile)
constexpr int WAVES_ROW = 8;                 // waves per block (split over k)
constexpr int TILE_F    = RPB * PITCH;       // 576 floats per buffer

// col pass tiling
constexpr int CPB    = 1024;                 // columns per block (256 thr x float4)
constexpr int RCH    = 256;                  // rows per chunk
constexpr int NROWCH = NDIM / RCH;           // 32 partial slabs

__global__ __launch_bounds__(256) void sk_init(float* __restrict__ f,
                                               float* __restrict__ g,
                                               float* __restrict__ gs) {
  int i = blockIdx.x * 256 + threadIdx.x;
  if (i < NDIM) { f[i] = 0.0f; g[i] = 0.0f; gs[i] = 0.0f; }
}

// ---- row pass: f[i] = eps*log(a[i]+1e-30) - eps*log( sum_j exp((g[j]-C[i,j])/eps) )
// gs[] is g pre-scaled by K2 (produced by sk_col_finish / sk_init).
__global__ __launch_bounds__(WAVES_ROW * 32)
void sk_row_pass(const float* __restrict__ cost, const float* __restrict__ a,
                 const float* __restrict__ gs, float* __restrict__ f) {
  __shared__ alignas(16) float tile[WAVES_ROW][2][TILE_F];  // double-buffered, wave-private
  __shared__ float psum[WAVES_ROW][RPB];

  const int tid  = threadIdx.x;
  const int wave = tid >> 5;
  const int lane = tid & 31;
  const int row0 = blockIdx.x * RPB;
  const int half = lane >> 4;       // A frag: 0 -> K{0,1}, 1 -> K{2,3}
  const int l15  = lane & 15;
  const int srow = lane >> 3;       // staging: 4 rows x 128B contiguous per instr
  const int scol = (lane & 7) << 2;

  const float* gbase = cost + (size_t)row0 * NDIM + scol;
  const unsigned lbuf0 = (unsigned)(uintptr_t)(&tile[wave][0][0]);
  const unsigned lbuf1 = (unsigned)(uintptr_t)(&tile[wave][1][0]);

  // async-stage one 16x32 tile (4 x b128 per lane-group => 4 wave instrs, 2KB)
  auto stage = [&](unsigned lb, int kc) {
#pragma unroll
    for (int s = 0; s < 4; ++s) {
      const int r = 4 * s + srow;
      const unsigned loff = lb + (unsigned)((r * PITCH + scol) * sizeof(float));
      const unsigned long long ga =
          (unsigned long long)(uintptr_t)(gbase + (size_t)r * NDIM + kc * TK);
      asm volatile("global_load_async_to_lds_b128 %0, %1, off"
                   :: "v"(loff), "v"(ga) : "memory");
    }
  };

  v8f acc = {};                     // 16x16 f32 accumulator (row sums, col-replicated)
  const v2f ones = {1.0f, 1.0f};    // all-ones B => D[m,n] = rowsum(A)[m]

  stage(lbuf0, wave);
  int bi = 0;
  for (int kc = wave; kc < NCHUNK; kc += WAVES_ROW, bi ^= 1) {
    const int kn = kc + WAVES_ROW;
    if (kn < NCHUNK) {
      stage(bi ? lbuf0 : lbuf1, kn);                    // prefetch next chunk
      asm volatile("s_wait_asynccnt 4" ::: "memory");   // current buffer resident
    } else {
      asm volatile("s_wait_asynccnt 0" ::: "memory");
    }
    const float* tl = &tile[wave][bi][0];
    // A fragment per ISA: lane L<16 -> row L, K{k,k+1}; lane L>=16 -> row L-16, K{k+2,k+3}
#pragma unroll
    for (int k4 = 0; k4 < TK / 4; ++k4) {
      const int kk = (k4 << 2) + (half << 1);
      const v2f c2 = *(const v2f*)(tl + l15 * PITCH + kk);
      const v2f g2 = *(const v2f*)(gs + kc * TK + kk);  // 8B broadcast, L2-hot
      v2f av;
      av.x = __builtin_amdgcn_exp2f(fmaf(c2.x, -K2, g2.x));
      av.y = __builtin_amdgcn_exp2f(fmaf(c2.y, -K2, g2.y));
      acc = __builtin_amdgcn_wmma_f32_16x16x4_f32(false, av, false, ones,
                                                  (short)0, acc, false, false);
    }
  }

  // lane 0 holds rows 0..7, lane 16 rows 8..15 (any D column = row sum)
  if (l15 == 0) {
#pragma unroll
    for (int i = 0; i < 8; ++i) psum[wave][half * 8 + i] = acc[i];
  }
  __syncthreads();
  if (tid < RPB) {
    float S = 0.0f;
#pragma unroll
    for (int w = 0; w < WAVES_ROW; ++w) S += psum[w][tid];
    const int row = row0 + tid;
    f[row] = EPS * (logf(a[row] + 1e-30f) - logf(S));
  }
}

// ---- col pass part 1: partial column sums of exp((f[i]-C[i,j])/eps)
__global__ __launch_bounds__(256)
void sk_col_partial(const float* __restrict__ cost, const float* __restrict__ f,
                    float* __restrict__ part) {
  const int col = blockIdx.x * CPB + threadIdx.x * 4;
  const int r0  = blockIdx.y * RCH;
  float4 acc = make_float4(0.f, 0.f, 0.f, 0.f);
#pragma unroll 4
  for (int r = r0; r < r0 + RCH; ++r) {
    const float fs = f[r] * K2;  // uniform -> scalar load
    const float4 c = *(const float4*)(cost + (size_t)r * NDIM + col);
    acc.x += __builtin_amdgcn_exp2f(fmaf(c.x, -K2, fs));
    acc.y += __builtin_amdgcn_exp2f(fmaf(c.y, -K2, fs));
    acc.z += __builtin_amdgcn_exp2f(fmaf(c.z, -K2, fs));
    acc.w += __builtin_amdgcn_exp2f(fmaf(c.w, -K2, fs));
  }
  *(float4*)(part + (size_t)blockIdx.y * NDIM + col) = acc;
}

// ---- col pass part 2: g[j] = eps*log(b[j]+1e-30) - eps*log(sum of partials)
// also emits gs[j] = g[j]*K2 for the next row pass.
__global__ __launch_bounds__(256)
void sk_col_finish(const float* __restrict__ part, const float* __restrict__ b,
                   float* __restrict__ g, float* __restrict__ gs) {
  const int j = blockIdx.x * 256 + threadIdx.x;
  float S = 0.f;
#pragma unroll
  for (int c = 0; c < NROWCH; ++c) S += part[(size_t)c * NDIM + j];
  const float gj = EPS * (logf(b[j] + 1e-30f) - logf(S));
  g[j]  = gj;
  gs[j] = gj * K2;
}

// ---- final plan: out = exp((f[i] + g[j] - C[i,j]) / eps)
__global__ __launch_bounds__(256)
void sk_plan(const float* __restrict__ cost, const float* __restrict__ f,
             const float* __restrict__ g, float* __restrict__ out) {
  const size_t idx4 = ((size_t)blockIdx.x * 256 + threadIdx.x) * 4;
  const int i = (int)(idx4 >> 13);         // blocks never straddle a row
  const int j = (int)(idx4 & (size_t)(NDIM - 1));
  const float  fi = f[i] * K2;
  const float4 gj = *(const float4*)(g + j);
  const float4 c  = *(const float4*)(cost + idx4);
  float4 o;
  o.x = __builtin_amdgcn_exp2f(fmaf(c.x, -K2, fmaf(gj.x, K2, fi)));
  o.y = __builtin_amdgcn_exp2f(fmaf(c.y, -K2, fmaf(gj.y, K2, fi)));
  o.z = __builtin_amdgcn_exp2f(fmaf(c.z, -K2, fmaf(gj.z, K2, fi)));
  o.w = __builtin_amdgcn_exp2f(fmaf(c.w, -K2, fmaf(gj.w, K2, fi)));
  *(float4*)(out + idx4) = o;
}

extern "C" void kernel_launch(void* const* d_in, const int* in_sizes, int n_in,
                              void* d_out, int out_size, void* d_ws, size_t ws_size,
                              hipStream_t stream) {
  const float* cost = (const float*)d_in[0];
  const float* a    = (const float*)d_in[1];
  const float* b    = (const float*)d_in[2];
  float* f    = (float*)d_ws;
  float* g    = f + NDIM;
  float* gs   = g + NDIM;                  // g pre-scaled by K2
  float* part = gs + NDIM;                 // NROWCH * NDIM floats (~1 MB)
  float* out  = (float*)d_out;

  sk_init<<<NDIM / 256, 256, 0, stream>>>(f, g, gs);
  for (int it = 0; it < 50; ++it) {
    sk_row_pass<<<NDIM / RPB, WAVES_ROW * 32, 0, stream>>>(cost, a, gs, f);
    sk_col_partial<<<dim3(NDIM / CPB, NROWCH), 256, 0, stream>>>(cost, f, part);
    sk_col_finish<<<NDIM / 256, 256, 0, stream>>>(part, b, g, gs);
  }
  sk_plan<<<(unsigned)(((size_t)NDIM * NDIM / 4) / 256), 256, 0, stream>>>(cost, f, g, out);
}